// IOU3DLoss_18674517803353
// MI455X (gfx1250) — compile-verified
//
#include <hip/hip_runtime.h>

// -----------------------------------------------------------------------------
// Rotated 3D IoU loss, MI455X (gfx1250, wave32).
//
// Roofline: N=500k pairs, 60 B in/pair = 30 MB total (fits in 192 MB L2);
// ~1.3us of HBM time at 23.3 TB/s => entirely VALU-bound. Optimization:
//  * Sutherland-Hodgman clipping of two convex quads (same intersection area
//    as the reference's 24x atan2 + argsort hull, ~4x fewer flops, no sort).
//  * Variable-length polygon kept in bank-conflict-free LDS [slot][tid];
//    rolling register reuse of the previous vertex halves DS loads; the final
//    clip pass streams directly into the shoelace accumulator (no 4th store).
//  * Weighted-mean reduction rides the CDNA5 matrix pipe:
//    V_WMMA_F32_16X16X4_F32 with A = lane values, B = ones gives
//    D[m][n] = v[m] + v[m+16]; 7 adds + one ds_swizzle (xor 16) complete the
//    wave sum. Deterministic block partials -> d_ws -> single-block finish.
// -----------------------------------------------------------------------------

#define BLOCK 256
#define NWAVES (BLOCK / 32)

typedef float v2f __attribute__((ext_vector_type(2)));
typedef float v8f __attribute__((ext_vector_type(8)));

// Wave32 sum via one f32 WMMA (A-side) + 7 adds + one ds_swizzle.
// A: lanes 0-15 put v in column K=0 (rows M=0..15), lanes 16-31 in K=2;
// B = ones => D[m][n] = v[m] + v[m+16] (independent of n).
// D layout: VGPR r holds row M=r (lanes 0-15) / M=r+8 (lanes 16-31), so the
// per-lane sum of the 8 D VGPRs yields sum(pairsum[0..7]) in the low half and
// sum(pairsum[8..15]) in the high half; one xor-16 swizzle adds the halves.
// Requires EXEC = all ones (call only from re-converged full blocks).
__device__ __forceinline__ float wave_sum32(float v) {
  v2f a; a[0] = v;    a[1] = 0.0f;   // A-matrix: values in K=0 / K=2 columns
  v2f b; b[0] = 1.0f; b[1] = 1.0f;   // B-matrix: all ones
  v8f c = {};
  c = __builtin_amdgcn_wmma_f32_16x16x4_f32(
      /*neg_a=*/false, a, /*neg_b=*/false, b,
      /*c_mod=*/(short)0, c, /*reuse_a=*/false, /*reuse_b=*/false);
  float s = ((c[0] + c[1]) + (c[2] + c[3])) + ((c[4] + c[5]) + (c[6] + c[7]));
  // ds_swizzle group-of-32: and=0x1f, or=0, xor=0x10 -> swap 16-lane halves.
  float o = __int_as_float(__builtin_amdgcn_ds_swizzle(__float_as_int(s), 0x401F));
  return s + o;                       // full wave sum in every lane
}

// One stored clip pass: clip polygon in (sx_in,sy_in)[0..n) by half-plane
// left-of edge (a -> a+e), write result into (sx_out,sy_out). Returns count.
__device__ __forceinline__ int clip_store(const float (*sx_in)[BLOCK],
                                          const float (*sy_in)[BLOCK],
                                          float (*sx_out)[BLOCK],
                                          float (*sy_out)[BLOCK],
                                          int n, int tid,
                                          float ax, float ay,
                                          float ex, float ey) {
  int m = 0;
  float P0x = sx_in[0][tid], P0y = sy_in[0][tid];
  float Px = P0x, Py = P0y;
  float dP = ex * (Py - ay) - ey * (Px - ax);
  for (int ii = 0; ii < n; ++ii) {
    float Qx, Qy;
    if (ii + 1 == n) { Qx = P0x; Qy = P0y; }
    else            { Qx = sx_in[ii + 1][tid]; Qy = sy_in[ii + 1][tid]; }
    float dQ = ex * (Qy - ay) - ey * (Qx - ax);
    bool inP = (dP >= 0.0f), inQ = (dQ >= 0.0f);
    if (inP) { sx_out[m][tid] = Px; sy_out[m][tid] = Py; ++m; }
    if (inP != inQ) {
      float tt = dP / (dP - dQ);
      sx_out[m][tid] = Px + tt * (Qx - Px);
      sy_out[m][tid] = Py + tt * (Qy - Py);
      ++m;
    }
    Px = Qx; Py = Qy; dP = dQ;
  }
  return m;
}

// Final clip pass fused with shoelace: stream emitted vertices into the area
// accumulator (track first & last emitted vertex, wrap at the end).
__device__ __forceinline__ float clip_area(const float (*sx_in)[BLOCK],
                                           const float (*sy_in)[BLOCK],
                                           int n, int tid,
                                           float ax, float ay,
                                           float ex, float ey) {
  float area2 = 0.0f;
  bool have_first = false;
  float Fx = 0.0f, Fy = 0.0f, Lx = 0.0f, Ly = 0.0f;
  float P0x = sx_in[0][tid], P0y = sy_in[0][tid];
  float Px = P0x, Py = P0y;
  float dP = ex * (Py - ay) - ey * (Px - ax);
  for (int ii = 0; ii < n; ++ii) {
    float Qx, Qy;
    if (ii + 1 == n) { Qx = P0x; Qy = P0y; }
    else            { Qx = sx_in[ii + 1][tid]; Qy = sy_in[ii + 1][tid]; }
    float dQ = ex * (Qy - ay) - ey * (Qx - ax);
    bool inP = (dP >= 0.0f), inQ = (dQ >= 0.0f);
    if (inP) {
      if (!have_first) { Fx = Px; Fy = Py; have_first = true; }
      else             { area2 += Lx * Py - Px * Ly; }
      Lx = Px; Ly = Py;
    }
    if (inP != inQ) {
      float tt = dP / (dP - dQ);
      float Ix = Px + tt * (Qx - Px);
      float Iy = Py + tt * (Qy - Py);
      if (!have_first) { Fx = Ix; Fy = Iy; have_first = true; }
      else             { area2 += Lx * Iy - Ix * Ly; }
      Lx = Ix; Ly = Iy;
    }
    Px = Qx; Py = Qy; dP = dQ;
  }
  if (have_first) area2 += Lx * Fy - Fx * Ly;   // close the polygon
  return 0.5f * fabsf(area2);
}

__global__ void __launch_bounds__(BLOCK)
iou3d_partial_kernel(const float* __restrict__ pred,
                     const float* __restrict__ target,
                     const float* __restrict__ weight,
                     float* __restrict__ partial, int N) {
  // Ping-pong polygon buffers, SoA [buf][slot][tid]; BLOCK%64==0 so the bank
  // index is tid%64 regardless of (divergent) slot => conflict-free.
  // Max verts after 3 clips = 7; 8 slots. 2*8*256*4*2 = 32 KB LDS/block.
  __shared__ float spx[2][8][BLOCK];
  __shared__ float spy[2][8][BLOCK];
  __shared__ float swsum[NWAVES];

  const int tid = threadIdx.x;
  const long long i = (long long)blockIdx.x * BLOCK + tid;

  float val = 0.0f;
  if (i < N) {
    float p[7], t[7];
#pragma unroll
    for (int k = 0; k < 7; ++k) {
      p[k] = pred[i * 7 + k];
      float tv = target[i * 7 + k];
      t[k] = (tv != tv) ? p[k] : tv;  // NaN in target -> fall back to pred
    }

    // box2corners on (x, y, w, h, yaw) = indices (0,1,3,4,6); CCW order.
    const float txs[4] = {0.5f, -0.5f, -0.5f, 0.5f};
    const float tys[4] = {0.5f, 0.5f, -0.5f, -0.5f};
    float s1, co1, s2, co2;
    __sincosf(p[6], &s1, &co1);
    __sincosf(t[6], &s2, &co2);
    float c2x[4], c2y[4];
#pragma unroll
    for (int k = 0; k < 4; ++k) {
      float xo = p[3] * txs[k], yo = p[4] * tys[k];
      spx[0][k][tid] = p[0] + co1 * xo - s1 * yo;   // box1 corners -> buf 0
      spy[0][k][tid] = p[1] + s1 * xo + co1 * yo;
      float xo2 = t[3] * txs[k], yo2 = t[4] * tys[k];
      c2x[k] = t[0] + co2 * xo2 - s2 * yo2;         // box2 corners (clipper)
      c2y[k] = t[1] + s2 * xo2 + co2 * yo2;
    }

    // Clip by box2 edges 0..2 (stored), fuse edge 3 with the shoelace area.
    float inter2d = 0.0f;
    int n = clip_store(spx[0], spy[0], spx[1], spy[1], 4, tid,
                       c2x[0], c2y[0], c2x[1] - c2x[0], c2y[1] - c2y[0]);
    if (n) {
      n = clip_store(spx[1], spy[1], spx[0], spy[0], n, tid,
                     c2x[1], c2y[1], c2x[2] - c2x[1], c2y[2] - c2y[1]);
      if (n) {
        n = clip_store(spx[0], spy[0], spx[1], spy[1], n, tid,
                       c2x[2], c2y[2], c2x[3] - c2x[2], c2y[3] - c2y[2]);
        if (n) {
          inter2d = clip_area(spx[1], spy[1], n, tid,
                              c2x[3], c2y[3], c2x[0] - c2x[3], c2y[0] - c2y[3]);
        }
      }
    }

    // z-overlap, volumes, IoU, weighted loss contribution.
    float zmax1 = p[2] + 0.5f * p[5], zmin1 = p[2] - 0.5f * p[5];
    float zmax2 = t[2] + 0.5f * t[5], zmin2 = t[2] - 0.5f * t[5];
    float zo = fmaxf(fminf(zmax1, zmax2) - fmaxf(zmin1, zmin2), 0.0f);
    float inter3d = inter2d * zo;
    float v1 = p[3] * p[4] * p[5];
    float v2 = t[3] * t[4] * t[5];
    float uni = v1 + v2 - inter3d;
    float iou = inter3d / uni;
    val = weight[i] * (1.0f - iou);
  }

  // Re-converged; full 256-thread block => EXEC all ones for WMMA.
  float wsum = wave_sum32(val);
  if ((tid & 31) == 0) swsum[tid >> 5] = wsum;
  __syncthreads();
  if (tid == 0) {
    float bsum = 0.0f;
#pragma unroll
    for (int k = 0; k < NWAVES; ++k) bsum += swsum[k];
    partial[blockIdx.x] = bsum;
  }
}

__global__ void __launch_bounds__(BLOCK)
reduce_final_kernel(const float* __restrict__ partial, int nparts,
                    float* __restrict__ out, float invN) {
  __shared__ float swsum[NWAVES];
  float acc = 0.0f;
  for (int k = threadIdx.x; k < nparts; k += BLOCK) acc += partial[k];
  float wsum = wave_sum32(acc);              // WMMA-based wave reduction
  if ((threadIdx.x & 31) == 0) swsum[threadIdx.x >> 5] = wsum;
  __syncthreads();
  if (threadIdx.x == 0) {
    float tot = 0.0f;
#pragma unroll
    for (int k = 0; k < NWAVES; ++k) tot += swsum[k];
    out[0] = tot * invN;                     // loss_weight * mean(...)
  }
}

extern "C" void kernel_launch(void* const* d_in, const int* in_sizes, int n_in,
                              void* d_out, int out_size, void* d_ws, size_t ws_size,
                              hipStream_t stream) {
  const float* pred   = (const float*)d_in[0];   // (N,7) f32
  const float* target = (const float*)d_in[1];   // (N,7) f32
  const float* weight = (const float*)d_in[2];   // (N,)  f32
  const int N = in_sizes[2];                     // weight element count == N

  float* partial = (float*)d_ws;                 // nblocks f32 partial sums
  const int nblocks = (N + BLOCK - 1) / BLOCK;

  iou3d_partial_kernel<<<nblocks, BLOCK, 0, stream>>>(pred, target, weight,
                                                      partial, N);
  reduce_final_kernel<<<1, BLOCK, 0, stream>>>(partial, nblocks,
                                               (float*)d_out,
                                               1.0f / (float)N);
}